// MinusAttention_30777735643477
// MI455X (gfx1250) — compile-verified
//
#include <hip/hip_runtime.h>
#include <hip/hip_bf16.h>
#include <math.h>

// Problem constants (from reference): B=4, L=S=2048, H=8, E=64
#define BC   4
#define LC   2048
#define SC   2048
#define HC   8
#define EC   64
#define NSEG 16
#define SEGLEN (SC / NSEG)      // 128
#define SCALE_F 0.125f          // 1/sqrt(64)

typedef __attribute__((ext_vector_type(2))) float v2f;
typedef __attribute__((ext_vector_type(8))) float v8f;

// ---------------------------------------------------------------------------
// Kernel 1: e[b,h,s] = exp(-scale * dot(keys[b,s,h,:], w))
// One wave (32 lanes) per (b,s,h) row; each lane covers 2 of the 64 elements.
// ---------------------------------------------------------------------------
__global__ __launch_bounds__(256) void ma_project_kernel(
    const float* __restrict__ keys, const float* __restrict__ w,
    float* __restrict__ e_arr)
{
    const int tid  = threadIdx.x;
    const int lane = tid & 31;
    const int wv   = tid >> 5;
    const long row = (long)blockIdx.x * 8 + wv;   // row = (b*S + s)*H + h

    const float* kp = keys + row * EC;
    float p = kp[lane] * w[lane] + kp[lane + 32] * w[lane + 32];
    #pragma unroll
    for (int off = 16; off >= 1; off >>= 1)
        p += __shfl_xor(p, off, 32);

    if (lane == 0) {
        const int  h  = (int)(row % HC);
        const long bs = row / HC;
        const int  s  = (int)(bs % SC);
        const int  b  = (int)(bs / SC);
        e_arr[((size_t)b * HC + h) * SC + s] = expf(-SCALE_F * p);
    }
}

// ---------------------------------------------------------------------------
// Kernel 2: per-(b,h,segment) partial sums: pden = sum e, pnum[d] = sum e*v
// 256 threads: 4 s-stripes x 64 d-channels, LDS reduction at the end.
// ---------------------------------------------------------------------------
__global__ __launch_bounds__(256) void ma_partial_kernel(
    const float* __restrict__ values, const float* __restrict__ e_arr,
    float* __restrict__ pden, float* __restrict__ pnum)
{
    __shared__ float lnum[256];
    __shared__ float lden[4];

    const int blk = blockIdx.x;          // (bh, seg)
    const int seg = blk % NSEG;
    const int bh  = blk / NSEG;
    const int b   = bh / HC, h = bh % HC;
    const int tid = threadIdx.x;
    const int d   = tid & 63;            // channel
    const int g   = tid >> 6;            // s-stripe 0..3

    const float* vbase = values + ((size_t)b * SC * HC + h) * EC
                       + (size_t)seg * SEGLEN * (HC * EC);
    const float* ebase = e_arr + (size_t)bh * SC + seg * SEGLEN;

    float num = 0.f, den = 0.f;
    #pragma unroll 4
    for (int i = g; i < SEGLEN; i += 4) {
        const float e = ebase[i];
        den += e;
        num += e * vbase[(size_t)i * (HC * EC) + d];
    }
    lnum[tid] = num;
    if (d == 0) lden[g] = den;
    __syncthreads();
    if (g == 0) {
        pnum[(size_t)blk * EC + d] =
            lnum[d] + lnum[d + 64] + lnum[d + 128] + lnum[d + 192];
        if (d == 0) pden[blk] = lden[0] + lden[1] + lden[2] + lden[3];
    }
}

// ---------------------------------------------------------------------------
// Kernel 3: causal prefix-scan + output.
// Block = (b,h,seg); 4 waves, each owns a 16-wide d-tile.
// Per 16-row chunk: T = LowerTri16 x (e .* V) via V_WMMA_F32_16X16X4_F32
// (4 chained K=4 steps, exact f32); denominator scan via wave shuffles;
// one v_rcp per row instead of precise-div chains; next-chunk prefetch.
// ---------------------------------------------------------------------------
__global__ __launch_bounds__(128) void ma_scan_out_kernel(
    const float* __restrict__ values, const float* __restrict__ e_arr,
    const float* __restrict__ pden, const float* __restrict__ pnum,
    float* __restrict__ out)
{
    const int blk  = blockIdx.x;         // (bh, seg)
    const int seg  = blk % NSEG;
    const int bh   = blk / NSEG;
    const int b    = bh / HC, h = bh % HC;
    const int tid  = threadIdx.x;
    const int lane = tid & 31;
    const int wv   = tid >> 5;           // 0..3 -> d-tile
    const int n    = lane & 15;          // column within tile / row for A
    const int hi   = lane >> 4;          // lane half selects K pair / M half
    const int d    = wv * 16 + n;

    // Base (exclusive prefix over preceding segments) — few dozen reads.
    float base_den = 0.f, base_num = 0.f;
    for (int p = 0; p < seg; ++p) {
        base_den += pden[bh * NSEG + p];
        base_num += pnum[(size_t)(bh * NSEG + p) * EC + d];
    }

    const int s0 = seg * SEGLEN;
    const float* vbase = values + ((size_t)b * SC * HC + h) * EC;
    const float* ebase = e_arr + (size_t)bh * SC;
    float* obase = out + ((size_t)b * LC * HC + h) * EC;

    for (int c = 0; c < SEGLEN / 16; ++c) {
        const int sc = s0 + c * 16;

        // Prefetch next chunk's 16 value rows (one line per lane) so the
        // fetch overlaps this chunk's WMMA chain. gfx1250 global_prefetch.
        if (c + 1 < SEGLEN / 16) {
            __builtin_prefetch(&vbase[(size_t)(sc + 16 + n) * (HC * EC) + d], 0, 0);
        }

        // e for this chunk: lane (&15) holds row n's e, duplicated per half.
        float e_reg = ebase[sc + n];

        // Inclusive 16-wide scan of e (each half computes identical scan;
        // shfl_up never crosses the half boundary for active adds).
        float cum = e_reg;
        #pragma unroll
        for (int off = 1; off < 16; off <<= 1) {
            float t = __shfl_up(cum, off, 32);
            if (n >= off) cum += t;
        }
        // Per-row reciprocal of the running denominator (v_rcp_f32, ~1 ulp):
        // lane n holds 1 / (base_den + cum_e[row n]).
        const float inv = __builtin_amdgcn_rcpf(base_den + cum);

        // Triangular matmul: acc[l,d] = sum_{j<=l} e_j * v[j,d]
        v8f acc = {};
        #pragma unroll
        for (int kb = 0; kb < 16; kb += 4) {
            const int j0 = kb + hi * 2;
            const float e0 = __shfl(e_reg, j0, 32);
            const float e1 = __shfl(e_reg, j0 + 1, 32);
            const float v0 = vbase[(size_t)(sc + j0)     * (HC * EC) + d];
            const float v1 = vbase[(size_t)(sc + j0 + 1) * (HC * EC) + d];
            v2f A, Bm;
            A.x  = (j0     <= n) ? 1.f : 0.f;   // A[m=n][k=j0]   (lower-tri ones)
            A.y  = (j0 + 1 <= n) ? 1.f : 0.f;   // A[m=n][k=j0+1]
            Bm.x = e0 * v0;                     // B[k=j0][n]   = e_j * v[j,d]
            Bm.y = e1 * v1;                     // B[k=j0+1][n]
            acc = __builtin_amdgcn_wmma_f32_16x16x4_f32(
                false, A, false, Bm, (short)0, acc, false, false);
        }

        // Full-chunk sums: row 15 lives in acc[7] of the hi half.
        const float chunk_num = __shfl(acc[7], 16 + n, 32);
        const float chunk_den = __shfl(cum, 15, 32);

        // Emit 8 rows per lane (C/D layout: M = r + 8*hi, N = n).
        #pragma unroll
        for (int r = 0; r < 8; ++r) {
            const int   m    = r + hi * 8;
            const float invm = __shfl(inv, m, 32);
            obase[(size_t)(sc + m) * (HC * EC) + d] = (base_num + acc[r]) * invm;
        }

        base_num += chunk_num;
        base_den += chunk_den;
    }
}

// ---------------------------------------------------------------------------
// Host-side launcher. Inputs: 0=queries (unused: cancels in softmax),
// 1=keys, 2=values, 3=w_score, 4=b_score (unused: cancels), 5=attn_mask.
// ---------------------------------------------------------------------------
extern "C" void kernel_launch(void* const* d_in, const int* in_sizes, int n_in,
                              void* d_out, int out_size, void* d_ws, size_t ws_size,
                              hipStream_t stream) {
    (void)in_sizes; (void)n_in; (void)out_size; (void)ws_size;
    const float* keys   = (const float*)d_in[1];
    const float* values = (const float*)d_in[2];
    const float* wsc    = (const float*)d_in[3];
    float* out = (float*)d_out;

    // Workspace layout (floats): e[B*H*S] | pden[B*H*NSEG] | pnum[B*H*NSEG*E]
    float* e_arr = (float*)d_ws;                           // 65536 f
    float* pden  = e_arr + (size_t)BC * HC * SC;           //   512 f
    float* pnum  = pden + (size_t)BC * HC * NSEG;          // 32768 f  (~386 KB total)

    const int rows = BC * SC * HC;                         // 65536
    ma_project_kernel<<<rows / 8, 256, 0, stream>>>(keys, wsc, e_arr);
    ma_partial_kernel<<<BC * HC * NSEG, 256, 0, stream>>>(values, e_arr, pden, pnum);
    ma_scan_out_kernel<<<BC * HC * NSEG, 128, 0, stream>>>(values, e_arr, pden, pnum, out);
}